// FlowNorm_29669634081225
// MI455X (gfx1250) — compile-verified
//
#include <hip/hip_runtime.h>
#include <math.h>

#define K_BINS 8
#define HID    32
#define NRAW   25          // 3K+1
#define BB     32
#define TT     4096
#define CC     512
#define ROWS   (BB*CC)     // 16384
#define PSTR   76          // floats per param row (16B-aligned bin records at +12)
#define NCHUNK 8
#define TCH    (TT/NCHUNK) // 512

typedef float v2f __attribute__((ext_vector_type(2)));
typedef float v8f __attribute__((ext_vector_type(8)));

// ---------------------------------------------------------------------------
// Kernel 1: raw moments Σx, Σx², Σx³, Σx⁴ per (b,c), chunked over T for
// parallelism; fixed-order partials (no atomics => deterministic).
// grid (B, NCHUNK), block 128; each thread owns 4 consecutive channels (b128
// loads, fully coalesced 2KB per block-row).
// ---------------------------------------------------------------------------
__global__ __launch_bounds__(128) void fn_moments(const float* __restrict__ x,
                                                  float* __restrict__ part) {
  const int b = blockIdx.x;
  const int chunk = blockIdx.y;
  const int c4 = threadIdx.x * 4;
  float s1x=0,s1y=0,s1z=0,s1w=0, s2x=0,s2y=0,s2z=0,s2w=0;
  float s3x=0,s3y=0,s3z=0,s3w=0, s4x=0,s4y=0,s4z=0,s4w=0;
  size_t base = ((size_t)b*TT + (size_t)chunk*TCH)*CC + c4;
  for (int tt = 0; tt < TCH; ++tt) {
    const float4 v = *(const float4*)(x + base + (size_t)tt*CC);
    __builtin_prefetch(x + base + (size_t)(tt+8)*CC, 0, 0);
    float qx=v.x*v.x, qy=v.y*v.y, qz=v.z*v.z, qw=v.w*v.w;
    s1x+=v.x; s1y+=v.y; s1z+=v.z; s1w+=v.w;
    s2x+=qx;  s2y+=qy;  s2z+=qz;  s2w+=qw;
    s3x+=qx*v.x; s3y+=qy*v.y; s3z+=qz*v.z; s3w+=qw*v.w;
    s4x+=qx*qx;  s4y+=qy*qy;  s4z+=qz*qz;  s4w+=qw*qw;
  }
  float4* prow = (float4*)part + (size_t)chunk*ROWS + (size_t)b*CC + c4;
  prow[0] = make_float4(s1x, s2x, s3x, s4x);
  prow[1] = make_float4(s1y, s2y, s3y, s4y);
  prow[2] = make_float4(s1z, s2z, s3z, s4z);
  prow[3] = make_float4(s1w, s2w, s3w, s4w);
}

// ---------------------------------------------------------------------------
// Kernel 2: finalize stats -> MLP via V_WMMA_F32_16X16X4_F32 -> spline params.
// 256 threads = 8 waves; each wave computes 16 rows. grid = ROWS/128.
// A layout (16x4 f32): VGPR0 = K0 (lanes 0-15) / K2 (lanes 16-31); VGPR1 = K1/K3.
// C/D layout: VGPR v = (M=v, N=lane) lanes 0-15 ; (M=v+8, N=lane-16) lanes 16-31.
// ---------------------------------------------------------------------------
__global__ __launch_bounds__(256) void fn_mlp_params(
    const float* __restrict__ part,
    const float* __restrict__ W1, const float* __restrict__ bias1,
    const float* __restrict__ W2, const float* __restrict__ bias2,
    float* __restrict__ params) {
  __shared__ float hbuf[8][16][33];   // hid tile per wave (padded: stride 33 -> conflict-free)
  __shared__ float rbuf[8][16][26];   // raw tile per wave
  const int tid  = threadIdx.x;
  const int wv   = tid >> 5;
  const int lane = tid & 31;
  const int lh   = lane & 15;
  const int hi   = lane >> 4;         // 0: lanes 0-15, 1: lanes 16-31
  const int r0   = blockIdx.x * 128 + wv * 16;
  const int row  = r0 + lh;           // lanes 16-31 duplicate (harmless)

  // --- reduce chunked partials (fixed order) ---
  float S1=0.f, S2=0.f, S3=0.f, S4=0.f;
  #pragma unroll
  for (int ch = 0; ch < NCHUNK; ++ch) {
    const float4 p = ((const float4*)part)[(size_t)ch*ROWS + row];
    S1 += p.x; S2 += p.y; S3 += p.z; S4 += p.w;
  }
  const float Tn = (float)TT;
  const float mu  = S1 / Tn;
  const float m2  = S2 / Tn, m3 = S3 / Tn, m4 = S4 / Tn;
  float var = (S2 - Tn*mu*mu) / (Tn - 1.0f);
  var = fmaxf(var, 0.0f);
  const float sig  = fmaxf(sqrtf(var), 1e-4f);
  const float isig = 1.0f / sig;
  const float mu2  = mu*mu;
  const float M3   = m3 - 3.0f*mu*m2 + 2.0f*mu*mu2;
  const float M4   = m4 - 4.0f*mu*m3 + 6.0f*mu2*m2 - 3.0f*mu2*mu2;
  const float is2  = isig*isig;
  const float skew  = M3 * is2 * isig;
  const float ekurt = M4 * is2 * is2 - 3.0f;

  if (!hi) {  // one writer per row
    float* P = params + (size_t)row * PSTR;
    P[8] = isig; P[9] = -mu * isig; P[10] = 0.0f; P[11] = 0.0f;
  }

  // --- layer 1: hid(16x32) = stats(16x2, K padded to 4) @ W1^T(2x32) ---
  v2f A;  A[0] = hi ? 0.0f : skew;             A[1] = hi ? 0.0f : ekurt;
  v2f B0; B0[0] = hi ? 0.0f : W1[lh*2+0];      B0[1] = hi ? 0.0f : W1[lh*2+1];
  v2f B1; B1[0] = hi ? 0.0f : W1[(16+lh)*2+0]; B1[1] = hi ? 0.0f : W1[(16+lh)*2+1];
  v8f z8 = {0.f,0.f,0.f,0.f,0.f,0.f,0.f,0.f};
  v8f h0 = __builtin_amdgcn_wmma_f32_16x16x4_f32(false, A, false, B0, (short)0, z8, false, false);
  v8f h1 = __builtin_amdgcn_wmma_f32_16x16x4_f32(false, A, false, B1, (short)0, z8, false, false);
  const float bL = bias1[lh], bH = bias1[16+lh];
  #pragma unroll
  for (int v = 0; v < 8; ++v) {
    hbuf[wv][v + hi*8][lh]      = fmaxf(h0[v] + bL, 0.0f);
    hbuf[wv][v + hi*8][16 + lh] = fmaxf(h1[v] + bH, 0.0f);
  }
  __syncthreads();

  // --- layer 2: raw(16x25) = hid(16x32) @ W2^T(32x25), 8 chained K=4 steps ---
  v8f c0 = {0.f,0.f,0.f,0.f,0.f,0.f,0.f,0.f};
  v8f c1 = {0.f,0.f,0.f,0.f,0.f,0.f,0.f,0.f};
  const bool okhi = (16 + lh) < NRAW;
  #pragma unroll
  for (int kt = 0; kt < 8; ++kt) {
    const int kk = kt*4 + hi*2;
    v2f A2;  A2[0]  = hbuf[wv][lh][kk];        A2[1]  = hbuf[wv][lh][kk+1];
    v2f Blo; Blo[0] = W2[lh*HID + kk];         Blo[1] = W2[lh*HID + kk + 1];
    v2f Bhi; Bhi[0] = okhi ? W2[(16+lh)*HID + kk]     : 0.0f;
             Bhi[1] = okhi ? W2[(16+lh)*HID + kk + 1] : 0.0f;
    c0 = __builtin_amdgcn_wmma_f32_16x16x4_f32(false, A2, false, Blo, (short)0, c0, false, false);
    c1 = __builtin_amdgcn_wmma_f32_16x16x4_f32(false, A2, false, Bhi, (short)0, c1, false, false);
  }
  const float b2L = bias2[lh];
  const float b2H = okhi ? bias2[16+lh] : 0.0f;
  #pragma unroll
  for (int v = 0; v < 8; ++v) {
    rbuf[wv][v + hi*8][lh] = c0[v] + b2L;
    if (okhi) rbuf[wv][v + hi*8][16+lh] = c1[v] + b2H;
  }
  __syncthreads();

  // --- per-row postprocess: softmax / softplus / cumsum -> param records ---
  if (tid < 128) {
    const int w2 = tid >> 4, M = tid & 15;
    float raw[NRAW];
    #pragma unroll
    for (int j = 0; j < NRAW; ++j) raw[j] = rbuf[w2][M][j];

    float wdt[8], hgt[8];
    {
      float mw = raw[0], mh = raw[8];
      #pragma unroll
      for (int j = 1; j < 8; ++j) { mw = fmaxf(mw, raw[j]); mh = fmaxf(mh, raw[8+j]); }
      float ew[8], eh[8], sw = 0.f, sh = 0.f;
      #pragma unroll
      for (int j = 0; j < 8; ++j) {
        ew[j] = expf(raw[j]   - mw); sw += ew[j];
        eh[j] = expf(raw[8+j] - mh); sh += eh[j];
      }
      const float cw = 10.0f / sw, chs = 10.0f / sh;   // softmax * 2*BOUND
      #pragma unroll
      for (int j = 0; j < 8; ++j) { wdt[j] = ew[j]*cw; hgt[j] = eh[j]*chs; }
    }
    float dv[9];
    #pragma unroll
    for (int j = 0; j < 9; ++j) {        // softplus + MIN_DERIV (stable form)
      const float r = raw[16+j];
      dv[j] = fmaxf(r, 0.0f) + log1pf(expf(-fabsf(r))) + 1e-3f;
    }
    float kx[9], ky[9];
    kx[0] = -5.0f; ky[0] = -5.0f;
    #pragma unroll
    for (int j = 1; j <= 8; ++j) { kx[j] = kx[j-1] + wdt[j-1]; ky[j] = ky[j-1] + hgt[j-1]; }

    float* P = params + (size_t)(blockIdx.x*128 + tid) * PSTR;
    #pragma unroll
    for (int j = 0; j < 8; ++j) P[j] = kx[j+1];        // search keys
    #pragma unroll
    for (int j = 0; j < 8; ++j) {                       // bin records (8 floats, 16B aligned)
      const float invw = 1.0f / fmaxf(wdt[j], 1e-8f);
      const float sj   = hgt[j] * invw;                 // s = h / max(w,1e-8)
      float* R = P + 12 + 8*j;
      R[0] = kx[j];        R[1] = invw;         R[2] = ky[j];
      R[3] = hgt[j] * sj;  R[4] = hgt[j]*dv[j];
      R[5] = sj;           R[6] = dv[j+1] + dv[j] - 2.0f*sj;  R[7] = 0.0f;
    }
  }
}

// ---------------------------------------------------------------------------
// Kernel 3: elementwise RQS transform. grid (B, C/128, NCHUNK), block 256.
// Params for 128 channels staged in LDS; search keys + affine in registers.
// ---------------------------------------------------------------------------
__global__ __launch_bounds__(256) void fn_transform(
    const float* __restrict__ x, const float* __restrict__ params,
    float* __restrict__ out) {
  __shared__ float sp[128 * PSTR];   // 38,912 B
  const int b  = blockIdx.x;
  const int c0 = blockIdx.y * 128;
  const int t0 = blockIdx.z * TCH;
  const float* gp = params + ((size_t)b * CC + c0) * PSTR;
  for (int k = threadIdx.x; k < 128 * PSTR; k += 256) sp[k] = gp[k];
  __syncthreads();

  const int cl = threadIdx.x & 127;
  const int tp = threadIdx.x >> 7;
  const float* myp = sp + cl * PSTR;
  float skx[8];
  #pragma unroll
  for (int j = 0; j < 8; ++j) skx[j] = myp[j];
  const float isig = myp[8], nms = myp[9];
  const float* bins = myp + 12;

  size_t base = ((size_t)b * TT + t0 + tp) * CC + c0 + cl;
  for (int tt = 0; tt < TCH; tt += 2) {
    const float xv = x[base + (size_t)tt * CC];
    __builtin_prefetch(x + base + (size_t)(tt + 16) * CC, 0, 0);
    const float z = fmaf(xv, isig, nms);           // (x - mu)/sig
    int cnt = 0;                                    // searchsorted(kx, z, 'right') - 1, clipped
    #pragma unroll
    for (int j = 0; j < 8; ++j) cnt += (z >= skx[j]) ? 1 : 0;
    const int idx = cnt > 7 ? 7 : cnt;
    const float4* rec = (const float4*)(bins + idx * 8);
    const float4 r0 = rec[0];                       // kx_l, inv_w, ky_l, h*s
    const float4 r1 = rec[1];                       // h*d,  s,     q,    pad
    float zeta = (z - r0.x) * r0.y;
    zeta = fminf(fmaxf(zeta, 0.0f), 1.0f);
    const float z2 = zeta * zeta;
    const float pp = zeta - z2;                     // zeta*(1-zeta)
    const float numer = fmaf(r0.w, z2, r1.x * pp);
    const float denom = fmaxf(fmaf(r1.z, pp, r1.y), 1e-8f);
    float rcp = __builtin_amdgcn_rcpf(denom);
    rcp = fmaf(fmaf(-denom, rcp, 1.0f), rcp, rcp);  // 1 Newton step
    const float res = fmaf(numer, rcp, r0.z);
    out[base + (size_t)tt * CC] = (fabsf(z) > 5.0f) ? z : res;
  }
}

// ---------------------------------------------------------------------------
extern "C" void kernel_launch(void* const* d_in, const int* in_sizes, int n_in,
                              void* d_out, int out_size, void* d_ws, size_t ws_size,
                              hipStream_t stream) {
  const float* x  = (const float*)d_in[0];
  const float* W1 = (const float*)d_in[1];
  const float* b1 = (const float*)d_in[2];
  const float* W2 = (const float*)d_in[3];
  const float* b2 = (const float*)d_in[4];
  float* out = (float*)d_out;

  float* part   = (float*)d_ws;                                            // 8*16384*4 f = 2 MB
  float* params = (float*)((char*)d_ws + (size_t)NCHUNK*ROWS*4*sizeof(float)); // 16384*76 f

  fn_moments   <<<dim3(BB, NCHUNK),      128, 0, stream>>>(x, part);
  fn_mlp_params<<<dim3(ROWS/128),        256, 0, stream>>>(part, W1, b1, W2, b2, params);
  fn_transform <<<dim3(BB, CC/128, NCHUNK), 256, 0, stream>>>(x, params, out);
}